// HOReIDLocalFeaturesComputer_54065048322089
// MI455X (gfx1250) — compile-verified
//
#include <hip/hip_runtime.h>
#include <hip/hip_bf16.h>

typedef __attribute__((ext_vector_type(2))) float v2f;
typedef __attribute__((ext_vector_type(8))) float v8f;

#define C_DIM  2048
#define K_DIM  13
#define HW     128
#define M_OUT  14   // 13 keypoint rows + 1 global row

// One workgroup = 4 waves; wave w handles C-tile (blockIdx.x*4 + w) of batch n = blockIdx.y.
// Fused: 13-keypoint weighted pooling (WMMA f32 16x16x4) + global mean+max, single pass over feat.
__global__ __launch_bounds__(128) void horeid_local_wmma_kernel(
    const float* __restrict__ feat,       // [N, C, H, W] = [N, 2048, 128]
    const float* __restrict__ scoremap,   // [N, 13, 128]
    float* __restrict__ out_fv)           // [N, 14, 2048]
{
    const int n    = blockIdx.y;
    const int tid  = threadIdx.x;
    const int lane = tid & 31;
    const int wave = tid >> 5;

    // Stage scoremap[n] as padded 16x128 A-matrix tile in LDS (rows 13..15 = 0).
    __shared__ float ldsA[16 * HW];       // 8 KB
    const float* sm = scoremap + (size_t)n * K_DIM * HW;
    #pragma unroll
    for (int i = 0; i < 4; ++i) {
        int idx4 = tid + i * 128;         // which float4 of the 512
        int m    = idx4 >> 5;             // 32 float4 per 128-float row
        int hw   = (idx4 & 31) * 4;
        float4 v = make_float4(0.f, 0.f, 0.f, 0.f);
        if (m < K_DIM) v = *(const float4*)(sm + m * HW + hw);
        *(float4*)(&ldsA[idx4 * 4]) = v;
    }
    __syncthreads();

    const int c0   = (blockIdx.x * 4 + wave) * 16;  // 16-channel tile base
    const int col  = lane & 15;                     // N-column / M-row owned by this lane
    const int koff = (lane >> 4) * 2;               // K split across lane halves

    // B: feat row for channel c0+col, stepping 4 hw-positions per WMMA.
    const float* pB = feat + ((size_t)n * C_DIM + (c0 + col)) * HW + koff;
    // A: scoremap row `col` in LDS, same K split.
    const float* pA = &ldsA[col * HW + koff];

    // Prefetch the back half of this lane's 512B channel row into cache.
    __builtin_prefetch(pB + 64, 0, 0);

    v8f acc = {0.f, 0.f, 0.f, 0.f, 0.f, 0.f, 0.f, 0.f};
    float gsum = 0.f;
    float gmax = -__builtin_inff();

    #pragma unroll
    for (int s = 0; s < 32; ++s) {
        v2f a = *(const v2f*)(pA + 4 * s);   // A 16x4 f32: lanes 0-15 K={0,1}, lanes 16-31 K={2,3}
        v2f b = *(const v2f*)(pB + 4 * s);   // B 4x16 f32: same K striping, N = lane&15
        // D = A x B + C  (16x16 f32 accumulator, 8 VGPRs)
        acc = __builtin_amdgcn_wmma_f32_16x16x4_f32(
            /*neg_a=*/false, a, /*neg_b=*/false, b,
            /*c_mod=*/(short)0, acc, /*reuse_a=*/false, /*reuse_b=*/false);
        // Global branch rides along on the already-loaded feat data.
        gsum += b.x + b.y;
        gmax  = fmaxf(gmax, fmaxf(b.x, b.y));
    }

    // Lanes l and l^16 together cover all 128 hw positions of channel c0+col.
    gsum += __shfl_xor(gsum, 16);
    gmax  = fmaxf(gmax, __shfl_xor(gmax, 16));
    const float glob = gsum * (1.0f / 128.0f) + gmax;

    // Store: D VGPR r -> row M = r + 8*(lane>>4); rows 13..15 are padding.
    float* outn = out_fv + (size_t)n * M_OUT * C_DIM;
    const int mbase = (lane >> 4) * 8;
    #pragma unroll
    for (int r = 0; r < 8; ++r) {
        int m = mbase + r;
        if (m < K_DIM) outn[(size_t)m * C_DIM + c0 + col] = acc[r];
    }
    if (lane < 16) outn[(size_t)K_DIM * C_DIM + c0 + col] = glob;
}

// conf: L1-normalize the 13 confidences; global slot = 1.0. One wave per batch item.
__global__ __launch_bounds__(32) void horeid_conf_kernel(
    const float* __restrict__ kc,        // [N, 13]
    float* __restrict__ out_conf)        // [N, 14]
{
    const int n    = blockIdx.x;
    const int lane = threadIdx.x;
    float v = (lane < K_DIM) ? kc[n * K_DIM + lane] : 0.f;
    float s = fabsf(v);
    #pragma unroll
    for (int off = 16; off >= 1; off >>= 1) s += __shfl_xor(s, off);
    const float denom = fmaxf(s, 1e-12f);
    if (lane < K_DIM)  out_conf[n * M_OUT + lane] = v / denom;
    if (lane == K_DIM) out_conf[n * M_OUT + lane] = 1.0f;
}

extern "C" void kernel_launch(void* const* d_in, const int* in_sizes, int n_in,
                              void* d_out, int out_size, void* d_ws, size_t ws_size,
                              hipStream_t stream) {
    const float* feat     = (const float*)d_in[0];
    const float* scoremap = (const float*)d_in[1];
    const float* kc       = (const float*)d_in[2];

    const int N = in_sizes[2] / K_DIM;           // 256
    float* out_fv   = (float*)d_out;             // [N, 14, 2048]
    float* out_conf = out_fv + (size_t)N * M_OUT * C_DIM;

    dim3 grid(C_DIM / (16 * 4), N);              // 32 tile-groups x N batches
    horeid_local_wmma_kernel<<<grid, 128, 0, stream>>>(feat, scoremap, out_fv);
    horeid_conf_kernel<<<N, 32, 0, stream>>>(kc, out_conf);
}